// VectorExpansionCalculator_64080912056552
// MI455X (gfx1250) — compile-verified
//
#include <hip/hip_runtime.h>

// CDNA5 (gfx1250) wave32 WMMA types
typedef float v2f __attribute__((ext_vector_type(2)));
typedef float v8f __attribute__((ext_vector_type(8)));

// Output block geometry: l=0..3, n_max = {8,7,6,5}, per-edge sizes {8,21,30,35} (total 94)
// LDS OUT layout per wave: OUT[edge(32)][n(8)][m(16) padded to 20]  (pad kills bank conflicts)
#define OUT_STRIDE_N 20
#define OUT_STRIDE_E 160
#define SH_STRIDE 20   // 16 values + pad, 16B-aligned rows
#define RB_STRIDE 12   // 8 values + pad, 16B-aligned rows

template <int S, int NK, int MB>
__device__ __forceinline__ void gather_store(const float* __restrict__ lw,
                                             float* __restrict__ dst,
                                             int lane, int e0w, int E, bool full) {
  // Wave-contiguous region: 32*S floats. 128-float (512B) fully-coalesced b128 stores.
  constexpr int TOT = 32 * S;
#pragma unroll
  for (int it = 0; it < (TOT + 127) / 128; ++it) {
    const int f = it * 128 + lane * 4;
    if (f < TOT) {
      float4 v;
      float* pv = reinterpret_cast<float*>(&v);
#pragma unroll
      for (int t = 0; t < 4; ++t) {
        const int ff = f + t;
        const int edge = ff / S;           // const-divisor -> mul/shift
        const int rem = ff - edge * S;
        const int row = rem / NK;
        const int n = rem - row * NK;
        pv[t] = lw[edge * OUT_STRIDE_E + n * OUT_STRIDE_N + (MB + row)];
      }
      if (full) {
        *reinterpret_cast<float4*>(dst + f) = v;   // global_store_b128, coalesced
      } else {
#pragma unroll
        for (int t = 0; t < 4; ++t) {
          if (e0w + (f + t) / S < E) dst[f + t] = pv[t];
        }
      }
    }
  }
}

__global__ __launch_bounds__(64) void vexp_wmma_kernel(const float* __restrict__ vec,
                                                       float* __restrict__ out,
                                                       int E) {
  const int lane = threadIdx.x & 31;
  const int wave = threadIdx.x >> 5;
  const int e0w = blockIdx.x * 64 + wave * 32;  // first edge this wave owns

  __shared__ float shS[2][32][SH_STRIDE];          // SH values,   5 KB
  __shared__ float rbS[2][32][RB_STRIDE];          // radial,      3 KB
  __shared__ float outS[2][32 * OUT_STRIDE_E];     // outer prods, 40 KB

  // ---------------- per-lane elementwise compute (1 edge / lane) ----------------
  int e = e0w + lane;
  int ec = (e < E) ? e : (E - 1);
  const float x = vec[3 * ec + 0];
  const float y = vec[3 * ec + 1];
  const float z = vec[3 * ec + 2];
  const float r = sqrtf(x * x + y * y + z * z);
  const float ir = 1.0f / (r + 1e-12f);
  const float nx = x * ir, ny = y * ir, nz = z * ir;

  // radial: R_n = fcut * sin((n+1)*pi*r/5) / (r+eps), fcut = (r<5) ? 0.5*(cos(pi r/5)+1) : 0
  const float th = r * 0.6283185307179586f;  // pi/5
  float s1, c1;
  __sincosf(th, &s1, &c1);
  const float fcut = (r < 5.0f) ? (0.5f * (c1 + 1.0f)) : 0.0f;
  const float w = fcut * ir;
  float rbv[8];
  {
    float sk = s1, ck = c1;
    rbv[0] = w * s1;
#pragma unroll
    for (int k = 1; k < 8; ++k) {  // angle-addition recurrence: sin/cos((k+1)theta)
      const float sn = sk * c1 + ck * s1;
      const float cn = ck * c1 - sk * s1;
      sk = sn; ck = cn;
      rbv[k] = w * sk;
    }
  }

  // real spherical harmonics l=0..3 on the unit vector (16 values)
  const float x2 = nx * nx, y2 = ny * ny, z2 = nz * nz;
  float shv[16];
  shv[0]  = 0.28209479177387814f;
  shv[1]  = 0.4886025119029199f * ny;
  shv[2]  = 0.4886025119029199f * nz;
  shv[3]  = 0.4886025119029199f * nx;
  shv[4]  = 1.0925484305920792f * nx * ny;
  shv[5]  = 1.0925484305920792f * ny * nz;
  shv[6]  = 0.31539156525252005f * (2.0f * z2 - x2 - y2);
  shv[7]  = 1.0925484305920792f * nx * nz;
  shv[8]  = 0.5462742152960396f * (x2 - y2);
  shv[9]  = 0.5900435899266435f * ny * (3.0f * x2 - y2);
  shv[10] = 2.890611442640554f  * nx * ny * nz;
  shv[11] = 0.4570457994644658f * ny * (4.0f * z2 - x2 - y2);
  shv[12] = 0.3731763325901154f * nz * (2.0f * z2 - 3.0f * x2 - 3.0f * y2);
  shv[13] = 0.4570457994644658f * nx * (4.0f * z2 - x2 - y2);
  shv[14] = 1.445305721320277f  * nz * (x2 - y2);
  shv[15] = 0.5900435899266435f * nx * (x2 - 3.0f * y2);

#pragma unroll
  for (int i = 0; i < 16; ++i) shS[wave][lane][i] = shv[i];
#pragma unroll
  for (int i = 0; i < 8; ++i) rbS[wave][lane][i] = rbv[i];
  __syncthreads();

  // ---------------- WMMA: 2 edges per v_wmma_f32_16x16x4_f32 ----------------
  // A(16x4): col K=0 = sh[edge a], K=1 = sh[edge b], K=2,3 = 0
  //   lanes 0-15 (M=lane): VGPR0=A[M][0], VGPR1=A[M][1]; lanes 16-31: K=2,3 -> 0
  // B(4x16): row0 = [rb_a | 0], row1 = [0 | rb_b], rows 2,3 = 0
  //   lanes 0-15 (N=lane): VGPR0=B[0][N], VGPR1=B[1][N]; lanes 16-31: rows 2,3 -> 0
  // D(16x16): D[m][n] = sh_a[m]*rb_a[n] (n<8) ; sh_b[m]*rb_b[n-8] (n>=8)
  const bool lo16 = lane < 16;
  const int l16 = lane & 15;
  const int l8 = lane & 7;
#pragma unroll
  for (int p = 0; p < 16; ++p) {
    const int ea = 2 * p, eb = 2 * p + 1;
    // Unconditional LDS loads (addresses valid for every lane; upper half-lanes
    // re-read lower-half addresses -> LDS broadcast). Zeroing happens as
    // v_cndmask afterwards, so no exec-mask branching around ds_load.
    const float a0 = shS[wave][ea][l16];
    const float a1 = shS[wave][eb][l16];
    const float b0 = rbS[wave][ea][l8];
    const float b1 = rbS[wave][eb][l8];
    v2f A;
    A.x = lo16 ? a0 : 0.0f;
    A.y = lo16 ? a1 : 0.0f;
    v2f B;
    B.x = (lane < 8) ? b0 : 0.0f;
    B.y = (lane >= 8 && lane < 16) ? b1 : 0.0f;
    v8f C = {};
    v8f D = __builtin_amdgcn_wmma_f32_16x16x4_f32(
        /*neg_a=*/false, A, /*neg_b=*/false, B,
        /*c_mod=*/(short)0, C, /*reuse_a=*/false, /*reuse_b=*/false);

    // scatter D to LDS: lane = N; VGPR v = M=v (+8 upper half-lanes)
    const int eo = lo16 ? ((l16 >= 8) ? eb : ea) : ((l16 >= 8) ? eb : ea);
    const int n8 = l16 & 7;
    const int mb = lo16 ? 0 : 8;
    float* dst = &outS[wave][eo * OUT_STRIDE_E + n8 * OUT_STRIDE_N + mb];
    float4 d0 = {D[0], D[1], D[2], D[3]};
    float4 d1 = {D[4], D[5], D[6], D[7]};
    *reinterpret_cast<float4*>(dst) = d0;       // ds_store_b128
    *reinterpret_cast<float4*>(dst + 4) = d1;   // ds_store_b128
  }
  __syncthreads();

  // ---------------- coalesced global stores (store-bandwidth is the roofline) ----
  const bool full = (e0w + 32) <= E;
  const float* lw = outS[wave];
  float* o0 = out;
  float* o1 = o0 + (size_t)E * 8;
  float* o2 = o1 + (size_t)E * 21;
  float* o3 = o2 + (size_t)E * 30;
  gather_store<8, 8, 0>(lw, o0 + (size_t)e0w * 8,  lane, e0w, E, full);  // l=0, m=0
  gather_store<21, 7, 1>(lw, o1 + (size_t)e0w * 21, lane, e0w, E, full); // l=1, m=1..3
  gather_store<30, 6, 4>(lw, o2 + (size_t)e0w * 30, lane, e0w, E, full); // l=2, m=4..8
  gather_store<35, 5, 9>(lw, o3 + (size_t)e0w * 35, lane, e0w, E, full); // l=3, m=9..15
}

extern "C" void kernel_launch(void* const* d_in, const int* in_sizes, int n_in,
                              void* d_out, int out_size, void* d_ws, size_t ws_size,
                              hipStream_t stream) {
  (void)n_in; (void)out_size; (void)d_ws; (void)ws_size;
  const float* vec = (const float*)d_in[0];
  float* out = (float*)d_out;
  const int E = in_sizes[0] / 3;
  const int blocks = (E + 63) / 64;
  vexp_wmma_kernel<<<blocks, 64, 0, stream>>>(vec, out, E);
}